// QuixerBlock_79250736545817
// MI455X (gfx1250) — compile-verified
//
#include <hip/hip_runtime.h>
#include <hip/hip_bf16.h>
#include <math.h>

// ---------------- problem constants ----------------
#define BSZ   64
#define SEQ   512
#define HID   128
#define KDIM  (SEQ * HID)      // 65536
#define ANC   512
#define NPHI  6
#define KCHUNKS 64
#define KC    (KDIM / KCHUNKS) // 1024

typedef float v2f __attribute__((ext_vector_type(2)));
typedef float v8f __attribute__((ext_vector_type(8)));

// ---------------- kernel 1: init logits with bias ----------------
__global__ void k_init_logits(const float* __restrict__ b_lcu, float* __restrict__ logits) {
    int i = blockIdx.x * blockDim.x + threadIdx.x;   // 64*512 elements
    if (i < BSZ * ANC) logits[i] = b_lcu[i & (ANC - 1)];
}

// ---------------- kernel 2: WMMA f32 GEMM  logits += x @ W_lcu^T ----------------
// One wave (32 threads) per block. Block (nTile, kChunk).
// Wave computes a 64x16 output strip (4 accumulators) over its K chunk,
// then atomically adds partials to logits.
__global__ __launch_bounds__(32) void k_gemm_lcu(const float* __restrict__ x,
                                                 const float* __restrict__ Wlcu,
                                                 float* __restrict__ logits) {
    const int n0    = blockIdx.x * 16;          // ancilla (N) tile base
    const int kbase = blockIdx.y * KC;          // K chunk base
    const int lane  = threadIdx.x;
    const int half  = lane >> 4;                // 0 or 1
    const int l16   = lane & 15;

    // A rows (x, row-major, stride KDIM): M = batch index
    const float* a0 = x + (size_t)(l16 +  0) * KDIM;
    const float* a1 = x + (size_t)(l16 + 16) * KDIM;
    const float* a2 = x + (size_t)(l16 + 32) * KDIM;
    const float* a3 = x + (size_t)(l16 + 48) * KDIM;
    // B row (W_lcu, row-major (ANC,KDIM)): N = ancilla index held by lane%16
    const float* br = Wlcu + (size_t)(n0 + l16) * KDIM;

    v8f acc0 = {}, acc1 = {}, acc2 = {}, acc3 = {};

    // ISA A 16x4 layout: lane L holds M=L%16; VGPR0/1 = K = 2*half, 2*half+1
    // -> contiguous float2 at column (k + 2*half). B mirrored.
    #pragma unroll 4
    for (int k = 0; k < KC; k += 4) {
        const int kk = kbase + k + 2 * half;
        v2f bv  = *(const v2f*)(br + kk);
        v2f av0 = *(const v2f*)(a0 + kk);
        v2f av1 = *(const v2f*)(a1 + kk);
        v2f av2 = *(const v2f*)(a2 + kk);
        v2f av3 = *(const v2f*)(a3 + kk);
        acc0 = __builtin_amdgcn_wmma_f32_16x16x4_f32(false, av0, false, bv, (short)0, acc0, false, false);
        acc1 = __builtin_amdgcn_wmma_f32_16x16x4_f32(false, av1, false, bv, (short)0, acc1, false, false);
        acc2 = __builtin_amdgcn_wmma_f32_16x16x4_f32(false, av2, false, bv, (short)0, acc2, false, false);
        acc3 = __builtin_amdgcn_wmma_f32_16x16x4_f32(false, av3, false, bv, (short)0, acc3, false, false);
    }

    // C/D layout: VGPR r, lane half h -> M = r + 8*h, N = lane%16
    const int ncol = n0 + l16;
    #pragma unroll
    for (int r = 0; r < 8; ++r) {
        int m = r + 8 * half;
        atomicAdd(&logits[(size_t)(m +  0) * ANC + ncol], acc0[r]);
        atomicAdd(&logits[(size_t)(m + 16) * ANC + ncol], acc1[r]);
        atomicAdd(&logits[(size_t)(m + 32) * ANC + ncol], acc2[r]);
        atomicAdd(&logits[(size_t)(m + 48) * ANC + ncol], acc3[r]);
    }
}

// ---------------- kernel 3: softmax + L2 normalize per batch row ----------------
__global__ __launch_bounds__(ANC) void k_softmax_l2(const float* __restrict__ logits,
                                                    float* __restrict__ amps) {
    const int b = blockIdx.x;
    const int t = threadIdx.x;
    __shared__ float red[ANC];

    float v = logits[(size_t)b * ANC + t];

    red[t] = v; __syncthreads();
    for (int s = ANC / 2; s > 0; s >>= 1) {
        if (t < s) red[t] = fmaxf(red[t], red[t + s]);
        __syncthreads();
    }
    float mx = red[0]; __syncthreads();

    float e = __expf(v - mx);
    red[t] = e; __syncthreads();
    for (int s = ANC / 2; s > 0; s >>= 1) {
        if (t < s) red[t] += red[t + s];
        __syncthreads();
    }
    float a = e / red[0]; __syncthreads();

    red[t] = a * a; __syncthreads();
    for (int s = ANC / 2; s > 0; s >>= 1) {
        if (t < s) red[t] += red[t + s];
        __syncthreads();
    }
    amps[(size_t)b * ANC + t] = a * __frsqrt_rn(red[0]);
}

// ---------------- kernel 4: per-(b,l) QSVT simulation -> probs(b,4) ----------------
struct c32 { float re, im; };
__device__ __forceinline__ c32 cmul(c32 a, c32 b) { return {a.re*b.re - a.im*b.im, a.re*b.im + a.im*b.re}; }
__device__ __forceinline__ c32 cmac(c32 acc, c32 a, c32 b) {
    acc.re = fmaf(a.re, b.re, fmaf(-a.im, b.im, acc.re));
    acc.im = fmaf(a.re, b.im, fmaf( a.im, b.re, acc.im));
    return acc;
}
__device__ __forceinline__ c32 cmacc(c32 acc, c32 a, c32 b) { // acc += conj(a)*b
    acc.re = fmaf(a.re, b.re, fmaf( a.im, b.im, acc.re));
    acc.im = fmaf(a.re, b.im, fmaf(-a.im, b.re, acc.im));
    return acc;
}

__global__ __launch_bounds__(SEQ) void k_qsvt(const float* __restrict__ x,
                                              const float* __restrict__ Wtok,
                                              const float* __restrict__ btok,
                                              const float* __restrict__ phi,
                                              const float* __restrict__ amps,
                                              float* __restrict__ probs) {
    const int b = blockIdx.x;
    const int l = threadIdx.x;

    // up = x[b,l,:] @ W_tok^T + b_tok
    const float* xr = x + ((size_t)b * SEQ + l) * HID;
    float p0 = btok[0], p1 = btok[1], p2 = btok[2], p3 = btok[3];
    #pragma unroll 4
    for (int h = 0; h < HID; ++h) {
        float xv = xr[h];
        p0 = fmaf(xv, Wtok[0 * HID + h], p0);
        p1 = fmaf(xv, Wtok[1 * HID + h], p1);
        p2 = fmaf(xv, Wtok[2 * HID + h], p2);
        p3 = fmaf(xv, Wtok[3 * HID + h], p3);
    }

    // build U = CNOT @ ((RZ(p1)RY(p0)) kron (RZ(p3)RY(p2)))
    float s0, c0, s2, c2, sp1, cp1, sp3, cp3;
    __sincosf(0.5f * p0, &s0, &c0);
    __sincosf(0.5f * p2, &s2, &c2);
    __sincosf(0.5f * p1, &sp1, &cp1);
    __sincosf(0.5f * p3, &sp3, &cp3);

    c32 ph0  = {cp1, -sp1};              // e^{-i p1/2}
    c32 ph0c = {cp1,  sp1};
    c32 ph1  = {cp3, -sp3};
    c32 ph1c = {cp3,  sp3};

    c32 A00 = { c0 * ph0.re,  c0 * ph0.im};
    c32 A01 = {-s0 * ph0.re, -s0 * ph0.im};
    c32 A10 = { s0 * ph0c.re, s0 * ph0c.im};
    c32 A11 = { c0 * ph0c.re, c0 * ph0c.im};

    c32 B00 = { c2 * ph1.re,  c2 * ph1.im};
    c32 B01 = {-s2 * ph1.re, -s2 * ph1.im};
    c32 B10 = { s2 * ph1c.re, s2 * ph1c.im};
    c32 B11 = { c2 * ph1c.re, c2 * ph1c.im};

    // Kron: K[2i+k][2j+m] = A[i][j]*B[k][m]; CNOT swaps rows 2 and 3
    c32 U[4][4];
    U[0][0] = cmul(A00, B00); U[0][1] = cmul(A00, B01); U[0][2] = cmul(A01, B00); U[0][3] = cmul(A01, B01);
    U[1][0] = cmul(A00, B10); U[1][1] = cmul(A00, B11); U[1][2] = cmul(A01, B10); U[1][3] = cmul(A01, B11);
    U[3][0] = cmul(A10, B00); U[3][1] = cmul(A10, B01); U[3][2] = cmul(A11, B00); U[3][3] = cmul(A11, B01); // row2 -> row3
    U[2][0] = cmul(A10, B10); U[2][1] = cmul(A10, B11); U[2][2] = cmul(A11, B10); U[2][3] = cmul(A11, B11); // row3 -> row2

    // state = (amp, 0, 0, 0)
    c32 s[4];
    s[0] = {amps[(size_t)b * ANC + l], 0.0f};
    s[1] = s[2] = s[3] = c32{0.0f, 0.0f};

    #pragma unroll
    for (int i = 0; i < NPHI; ++i) {
        float sh, ch;
        __sincosf(0.5f * phi[i], &sh, &ch);
        c32 dp = {ch, -sh};   // Z = +1 (data idx 0,1)
        c32 dm = {ch,  sh};   // Z = -1 (data idx 2,3)
        s[0] = cmul(s[0], dp); s[1] = cmul(s[1], dp);
        s[2] = cmul(s[2], dm); s[3] = cmul(s[3], dm);

        c32 t[4];
        if ((i & 1) == 0) {                 // U @ s
            #pragma unroll
            for (int r = 0; r < 4; ++r) {
                c32 acc = {0.0f, 0.0f};
                #pragma unroll
                for (int cCol = 0; cCol < 4; ++cCol) acc = cmac(acc, U[r][cCol], s[cCol]);
                t[r] = acc;
            }
        } else {                            // U^dagger @ s : t_r = sum_j conj(U[j][r]) s[j]
            #pragma unroll
            for (int r = 0; r < 4; ++r) {
                c32 acc = {0.0f, 0.0f};
                #pragma unroll
                for (int j = 0; j < 4; ++j) acc = cmacc(acc, U[j][r], s[j]);
                t[r] = acc;
            }
        }
        #pragma unroll
        for (int r = 0; r < 4; ++r) s[r] = t[r];
    }

    float pr[4];
    #pragma unroll
    for (int j = 0; j < 4; ++j) pr[j] = s[j].re * s[j].re + s[j].im * s[j].im;

    __shared__ float red[4][SEQ];
    #pragma unroll
    for (int j = 0; j < 4; ++j) red[j][l] = pr[j];
    __syncthreads();
    for (int st = SEQ / 2; st > 0; st >>= 1) {
        if (l < st) {
            #pragma unroll
            for (int j = 0; j < 4; ++j) red[j][l] += red[j][l + st];
        }
        __syncthreads();
    }
    if (l == 0) {
        #pragma unroll
        for (int j = 0; j < 4; ++j) probs[b * 4 + j] = red[j][0];
    }
}

// ---------------- kernel 5: readout + broadcast (float4 stores) ----------------
__global__ void k_readout(const float* __restrict__ probs,
                          const float* __restrict__ Wro,
                          const float* __restrict__ bro,
                          float4* __restrict__ out) {
    const int total4 = BSZ * SEQ * HID / 4;
    int i = blockIdx.x * blockDim.x + threadIdx.x;
    if (i >= total4) return;
    int h  = (i & 31) * 4;            // HID/4 = 32 quads per row
    int b  = i >> 14;                 // / (SEQ * HID/4) = / 16384
    float q0 = probs[b * 4 + 0], q1 = probs[b * 4 + 1];
    float q2 = probs[b * 4 + 2], q3 = probs[b * 4 + 3];
    float z0 = q0 + q1 - q2 - q3;
    float z1 = q0 - q1 + q2 - q3;
    float4 v;
    v.x = fmaf(z0, Wro[(h + 0) * 2], fmaf(z1, Wro[(h + 0) * 2 + 1], bro[h + 0]));
    v.y = fmaf(z0, Wro[(h + 1) * 2], fmaf(z1, Wro[(h + 1) * 2 + 1], bro[h + 1]));
    v.z = fmaf(z0, Wro[(h + 2) * 2], fmaf(z1, Wro[(h + 2) * 2 + 1], bro[h + 2]));
    v.w = fmaf(z0, Wro[(h + 3) * 2], fmaf(z1, Wro[(h + 3) * 2 + 1], bro[h + 3]));
    out[i] = v;
}

// ---------------- launcher ----------------
extern "C" void kernel_launch(void* const* d_in, const int* in_sizes, int n_in,
                              void* d_out, int out_size, void* d_ws, size_t ws_size,
                              hipStream_t stream) {
    const float* x     = (const float*)d_in[0];
    const float* Wtok  = (const float*)d_in[1];
    const float* btok  = (const float*)d_in[2];
    const float* Wlcu  = (const float*)d_in[3];
    const float* blcu  = (const float*)d_in[4];
    const float* phi   = (const float*)d_in[5];
    const float* Wro   = (const float*)d_in[6];
    const float* bro   = (const float*)d_in[7];
    float* out = (float*)d_out;

    // workspace layout
    float* logits = (float*)d_ws;                         // 64*512 f32
    float* amps   = logits + BSZ * ANC;                   // 64*512 f32
    float* probs  = amps + BSZ * ANC;                     // 64*4   f32

    // 1) logits = b_lcu (broadcast)
    k_init_logits<<<(BSZ * ANC + 255) / 256, 256, 0, stream>>>(blcu, logits);

    // 2) logits += x_flat @ W_lcu^T  (WMMA f32, K-split atomics)
    dim3 ggrid(ANC / 16, KCHUNKS);
    k_gemm_lcu<<<ggrid, 32, 0, stream>>>(x, Wlcu, logits);

    // 3) amps = l2norm(softmax(logits))
    k_softmax_l2<<<BSZ, ANC, 0, stream>>>(logits, amps);

    // 4) QSVT per-(b,l) simulation -> probs
    k_qsvt<<<BSZ, SEQ, 0, stream>>>(x, Wtok, btok, phi, amps, probs);

    // 5) readout + broadcast to (B,S,H)
    int total4 = BSZ * SEQ * HID / 4;
    k_readout<<<(total4 + 255) / 256, 256, 0, stream>>>(probs, Wro, bro, (float4*)out);
}